// NNUE_59098749993087
// MI455X (gfx1250) — compile-verified
//
#include <hip/hip_runtime.h>
#include <hip/hip_bf16.h>

#define IN_FEATS 40960
#define H1 256
#define BATCH 16384

typedef float v2f __attribute__((ext_vector_type(2)));
typedef float v8f __attribute__((ext_vector_type(8)));

__device__ __forceinline__ float clip01(float x) { return fminf(fmaxf(x, 0.0f), 1.0f); }

// ---------------------------------------------------------------------------
// K0: LDS-tiled transpose W1 [H1, IN_FEATS] -> W1T [IN_FEATS, H1]
// so each feature column becomes 1KB contiguous for coalesced wave gathers.
// ---------------------------------------------------------------------------
__global__ void transpose_w1(const float* __restrict__ W1, float* __restrict__ W1T) {
  __shared__ float tile[32][33];
  int f0 = blockIdx.x * 32;
  int c0 = blockIdx.y * 32;
  int tx = threadIdx.x, ty = threadIdx.y;
#pragma unroll
  for (int i = 0; i < 4; ++i) {
    int c = c0 + ty + i * 8;
    tile[ty + i * 8][tx] = W1[(size_t)c * IN_FEATS + (f0 + tx)];
  }
  __syncthreads();
#pragma unroll
  for (int i = 0; i < 4; ++i) {
    int f = f0 + ty + i * 8;
    W1T[(size_t)f * H1 + (c0 + tx)] = tile[tx][ty + i * 8];
  }
}

// ---------------------------------------------------------------------------
// K1: zero the per-row counters (curW and curB are contiguous in d_ws)
// ---------------------------------------------------------------------------
__global__ void zero_i32(int* __restrict__ p, int n) {
  int i = blockIdx.x * 256 + threadIdx.x;
  if (i < n) p[i] = 0;
}

// ---------------------------------------------------------------------------
// K2: per-row histogram of both sides (4-byte atomics only)
// ---------------------------------------------------------------------------
__global__ void hist_kernel(const int* __restrict__ wi, const int* __restrict__ bi,
                            int* __restrict__ curW, int* __restrict__ curB, int NF) {
  int i = blockIdx.x * 256 + threadIdx.x;
  if (i < NF) {
    atomicAdd(&curW[wi[i]], 1);
  } else if (i < 2 * NF) {
    atomicAdd(&curB[bi[i - NF]], 1);
  }
}

// ---------------------------------------------------------------------------
// K3: block-wide exclusive scan (1024 threads x 16 rows each) per side.
// Writes row offsets and re-seeds the cursor array for the scatter pass.
// ---------------------------------------------------------------------------
__global__ void scan_kernel(int* __restrict__ curW, int* __restrict__ offsW,
                            int* __restrict__ curB, int* __restrict__ offsB) {
  int side = blockIdx.x;
  int* cnt  = side ? curB : curW;
  int* offs = side ? offsB : offsW;
  int t = threadIdx.x;
  int base = t * 16;
  int local[16];
  int sum = 0;
#pragma unroll
  for (int i = 0; i < 16; ++i) { local[i] = cnt[base + i]; sum += local[i]; }
  __shared__ int s[1024];
  s[t] = sum;
  __syncthreads();
  for (int o = 1; o < 1024; o <<= 1) {
    int v = (t >= o) ? s[t - o] : 0;
    __syncthreads();
    s[t] += v;
    __syncthreads();
  }
  int run = s[t] - sum;  // exclusive prefix
#pragma unroll
  for (int i = 0; i < 16; ++i) {
    offs[base + i] = run;
    cnt[base + i]  = run;  // cursor for scatter
    run += local[i];
  }
  if (t == 1023) offs[BATCH] = run;
}

// ---------------------------------------------------------------------------
// K4: scatter feature ids into CSR buckets
// ---------------------------------------------------------------------------
__global__ void scatter_kernel(const int* __restrict__ wf, const int* __restrict__ wi,
                               const int* __restrict__ bf, const int* __restrict__ bi,
                               int* __restrict__ curW, int* __restrict__ curB,
                               int* __restrict__ sW, int* __restrict__ sB, int NF) {
  int i = blockIdx.x * 256 + threadIdx.x;
  if (i < NF) {
    int r = wi[i];
    int pos = atomicAdd(&curW[r], 1);
    sW[pos] = wf[i];
  } else if (i < 2 * NF) {
    int j = i - NF;
    int r = bi[j];
    int pos = atomicAdd(&curB[r], 1);
    sB[pos] = bf[j];
  }
}

// ---------------------------------------------------------------------------
// K5: one wave32 per (row, side): gather + register accumulate 256 floats
// (8 per lane), fused bias + hardtanh, coalesced 512B loads/stores,
// global_prefetch of the next feature column.
// ---------------------------------------------------------------------------
__global__ void accum_rows(const float* __restrict__ W1T,
                           const int* __restrict__ offsW, const int* __restrict__ sortedW,
                           const int* __restrict__ offsB, const int* __restrict__ sortedB,
                           const float* __restrict__ b1, float* __restrict__ acc) {
  int gwave = (blockIdx.x * 256 + threadIdx.x) >> 5;
  int lane  = threadIdx.x & 31;
  int side  = gwave >> 14;            // 0 = white, 1 = black
  int row   = gwave & (BATCH - 1);
  const int* offs   = side ? offsB : offsW;
  const int* sorted = side ? sortedB : sortedW;
  int beg = offs[row], end = offs[row + 1];

  float a0x = 0.f, a0y = 0.f, a0z = 0.f, a0w = 0.f;
  float a1x = 0.f, a1y = 0.f, a1z = 0.f, a1w = 0.f;

  for (int j = beg; j < end; ++j) {
    int f = sorted[j];
    const float* col = W1T + (size_t)f * H1;
    if (j + 1 < end) {  // wave-uniform branch
      const float* nxt = W1T + (size_t)sorted[j + 1] * H1;
      __builtin_prefetch(nxt + 4 * lane, 0, 0);
      __builtin_prefetch(nxt + 128 + 4 * lane, 0, 0);
    }
    float4 x0 = *(const float4*)(col + 4 * lane);          // cols 4l..4l+3
    float4 x1 = *(const float4*)(col + 128 + 4 * lane);    // cols 128+4l..
    a0x += x0.x; a0y += x0.y; a0z += x0.z; a0w += x0.w;
    a1x += x1.x; a1y += x1.y; a1z += x1.z; a1w += x1.w;
  }

  float4 bb0 = *(const float4*)(b1 + 4 * lane);
  float4 bb1 = *(const float4*)(b1 + 128 + 4 * lane);
  float4 r0, r1;
  r0.x = clip01(a0x + bb0.x); r0.y = clip01(a0y + bb0.y);
  r0.z = clip01(a0z + bb0.z); r0.w = clip01(a0w + bb0.w);
  r1.x = clip01(a1x + bb1.x); r1.y = clip01(a1y + bb1.y);
  r1.z = clip01(a1z + bb1.z); r1.w = clip01(a1w + bb1.w);

  float* dst = acc + (size_t)row * 512 + side * 256;
  *(float4*)(dst + 4 * lane) = r0;
  *(float4*)(dst + 128 + 4 * lane) = r1;
}

// ---------------------------------------------------------------------------
// K6: layers 2+3. One wave per 16 batch rows. Exact f32 WMMA 16x16x4:
//   A frag (16x4):  lane l holds M=l&15, K-group l>>4, 2 consecutive f32
//   B frag (4x16):  lane l holds N=l&15, K-group l>>4, 2 consecutive f32
//   C/D (16x16):    vgpr v holds M=v+8*(l>=16), N=l&15
// 128 K-steps over the 512-wide clipped concat activation; 2 N-tiles.
// Layer 3 (32 -> 1) via shfl_xor butterfly within each 16-lane half.
// ---------------------------------------------------------------------------
__global__ void head_wmma(const float* __restrict__ acc, const float* __restrict__ W2,
                          const float* __restrict__ b2, const float* __restrict__ W3,
                          const float* __restrict__ b3, float* __restrict__ out) {
  int lane = threadIdx.x & 31;
  int wib  = threadIdx.x >> 5;
  int tile = blockIdx.x * 8 + wib;
  int row0 = tile * 16;
  int m  = lane & 15;
  int kg = lane >> 4;

  const float* arow  = acc + (size_t)(row0 + m) * 512 + 2 * kg;
  const float* brow0 = W2 + (size_t)m * 512 + 2 * kg;          // N = m, n0 = 0
  const float* brow1 = W2 + (size_t)(m + 16) * 512 + 2 * kg;   // n0 = 16

  v8f c0 = {0.f, 0.f, 0.f, 0.f, 0.f, 0.f, 0.f, 0.f};
  v8f c1 = {0.f, 0.f, 0.f, 0.f, 0.f, 0.f, 0.f, 0.f};

#pragma unroll 4
  for (int s = 0; s < 128; ++s) {
    v2f a  = *(const v2f*)(arow + 4 * s);
    v2f bA = *(const v2f*)(brow0 + 4 * s);
    v2f bB = *(const v2f*)(brow1 + 4 * s);
    c0 = __builtin_amdgcn_wmma_f32_16x16x4_f32(false, a, false, bA, (short)0, c0,
                                               false, false);
    c1 = __builtin_amdgcn_wmma_f32_16x16x4_f32(false, a, false, bB, (short)0, c1,
                                               false, false);
  }

  float w3a = W3[m], w3b = W3[m + 16];
  float b2a = b2[m], b2b = b2[m + 16];
  float p[8];
#pragma unroll
  for (int v = 0; v < 8; ++v) {
    float x0 = clip01(c0[v] + b2a);
    float x1 = clip01(c1[v] + b2b);
    float t = x0 * w3a + x1 * w3b;  // partial over N for row M=v+8*kg
    t += __shfl_xor(t, 1, 32);
    t += __shfl_xor(t, 2, 32);
    t += __shfl_xor(t, 4, 32);
    t += __shfl_xor(t, 8, 32);
    p[v] = t;
  }
  if (m == 0) {
    float bias3 = b3[0];
    int mb = row0 + 8 * kg;
#pragma unroll
    for (int v = 0; v < 8; ++v) out[mb + v] = p[v] + bias3;
  }
}

// ---------------------------------------------------------------------------
extern "C" void kernel_launch(void* const* d_in, const int* in_sizes, int n_in,
                              void* d_out, int out_size, void* d_ws, size_t ws_size,
                              hipStream_t stream) {
  const int*   wf = (const int*)d_in[1];
  const int*   wi = (const int*)d_in[2];
  const int*   bf = (const int*)d_in[3];
  const int*   bi = (const int*)d_in[4];
  const float* W1 = (const float*)d_in[5];
  const float* b1 = (const float*)d_in[6];
  const float* W2 = (const float*)d_in[7];
  const float* b2 = (const float*)d_in[8];
  const float* W3 = (const float*)d_in[9];
  const float* b3 = (const float*)d_in[10];
  float* out = (float*)d_out;
  const int NF = in_sizes[1];  // 491520

  // Workspace carve-up (256B aligned regions, ~80 MB total)
  char* base = (char*)d_ws;
  size_t off = 0;
  auto take = [&](size_t bytes) -> char* {
    char* p = base + off;
    off = (off + bytes + 255) & ~(size_t)255;
    return p;
  };
  float* W1T     = (float*)take((size_t)IN_FEATS * H1 * 4);
  float* accbuf  = (float*)take((size_t)BATCH * 512 * 4);
  int*   offsW   = (int*)take((size_t)(BATCH + 1) * 4);
  int*   offsB   = (int*)take((size_t)(BATCH + 1) * 4);
  int*   curW    = (int*)take((size_t)BATCH * 4);  // 65536B -> curB contiguous
  int*   curB    = (int*)take((size_t)BATCH * 4);
  int*   sortedW = (int*)take((size_t)NF * 4);
  int*   sortedB = (int*)take((size_t)NF * 4);
  (void)ws_size; (void)n_in; (void)out_size;

  const int twoNF = 2 * NF;

  transpose_w1<<<dim3(IN_FEATS / 32, H1 / 32), dim3(32, 8), 0, stream>>>(W1, W1T);
  zero_i32<<<(2 * BATCH + 255) / 256, 256, 0, stream>>>(curW, 2 * BATCH);
  hist_kernel<<<(twoNF + 255) / 256, 256, 0, stream>>>(wi, bi, curW, curB, NF);
  scan_kernel<<<2, 1024, 0, stream>>>(curW, offsW, curB, offsB);
  scatter_kernel<<<(twoNF + 255) / 256, 256, 0, stream>>>(wf, wi, bf, bi, curW, curB,
                                                          sortedW, sortedB, NF);
  accum_rows<<<(2 * BATCH) / 8, 256, 0, stream>>>(W1T, offsW, sortedW, offsB, sortedB,
                                                  b1, accbuf);
  head_wmma<<<BATCH / 16 / 8, 256, 0, stream>>>(accbuf, W2, b2, W3, b3, out);
}